// ScalableAlgebraicTransformer_78572131713325
// MI455X (gfx1250) — compile-verified
//
#include <hip/hip_runtime.h>
#include <hip/hip_bf16.h>

typedef float v2f __attribute__((ext_vector_type(2)));
typedef float v8f __attribute__((ext_vector_type(8)));

#define BATCH   512
#define DTOT    86016          // 64^2 + 128^2 + 256^2
#define NSTRIP  28             // 16-row strips per sample: 4 + 8 + 16
#define NPART   (NSTRIP * 8)   // partial-moment slots per sample (one per wave)
#define EPSLN   1e-5f

// ---------------------------------------------------------------------------
// One block = one (sample b, rep-block r, 16-row strip mt).
// 256 threads stage the 16 x d A-strip into LDS with coalesced float4 loads
// (each A element fetched from global exactly once per strip), then the 8
// waves sweep the d/16 N-tiles, each tile a K-loop of V_WMMA_F32_16X16X4_F32.
// LDS rows padded +4 floats: (d+4) % 64 == 4 for d in {64,128,256}, so the
// per-lane A-fragment ds_load_b64 hits all 64 banks exactly once.
// A-fragment layout (ISA 16x4 MxK): lanes 0-15 -> K=k0,k0+1; lanes 16-31 ->
// K=k0+2,k0+3. B mirrored, row-striped across lanes. C/D: VGPR j -> M=j(+8).
// ---------------------------------------------------------------------------
__global__ __launch_bounds__(256) void sat_mm(
    const float* __restrict__ rep0, const float* __restrict__ rep1,
    const float* __restrict__ rep2,
    const int*  __restrict__ transitions, const int* __restrict__ sequence,
    const float* __restrict__ X, float* __restrict__ Y,
    float* __restrict__ partSum, float* __restrict__ partSq,
    int firstLayer)
{
    __shared__ float As[16 * 260];                 // max strip: 16 x (256+4)

    const int tid  = threadIdx.x;
    const int lane = tid & 31;
    const int w    = tid >> 5;                     // wave id 0..7
    const int bid  = blockIdx.x;
    const int b    = bid / NSTRIP;
    const int s    = bid - b * NSTRIP;

    int r, d, off, mt;
    if (s < 4)       { r = 0; d = 64;  off = 0;     mt = s;      }
    else if (s < 12) { r = 1; d = 128; off = 4096;  mt = s - 4;  }
    else             { r = 2; d = 256; off = 20480; mt = s - 12; }
    const int pad    = d + 4;
    const int ntiles = d >> 4;

    const float* rep  = (r == 0) ? rep0 : ((r == 1) ? rep1 : rep2);
    const int    t    = transitions[sequence[b]];
    const float* Bmat = rep + (size_t)t * d * d;
    // Layer 0: every sample starts from rep[0] (identity element) -- read the
    // A strip and residual straight from the rep tensor, never materialize x0.
    const float* Ablk   = firstLayer ? rep : (X + (size_t)b * DTOT + off);
    const float* Astrip = Ablk + (size_t)mt * 16 * d;

    // ---- cooperative coalesced stage: global -> LDS (padded rows) ----
    const int nquad = (16 * d) >> 2;               // 256 / 512 / 1024 float4s
    for (int q4 = tid; q4 < nquad; q4 += 256) {
        const int e   = q4 << 2;
        const int row = e / d;
        const int col = e - row * d;
        float4 v = *(const float4*)(Astrip + (size_t)row * d + col);
        *(float4*)(&As[row * pad + col]) = v;
    }
    __syncthreads();

    const int mlo   = lane & 15;
    const int khalf = (lane >> 4) << 1;            // 0 (lanes 0-15) / 2 (16-31)
    const float* aLds = &As[mlo * pad + khalf];
    float s1 = 0.f, s2 = 0.f;

    for (int nt = w; nt < ntiles; nt += 8) {       // wave-uniform trip count
        const int    ncol = (nt << 4) + mlo;
        const float* bcol = Bmat + ncol;
        v8f acc = {};
        for (int k0 = 0; k0 < d; k0 += 4) {
            v2f a = *(const v2f*)(aLds + k0);              // LDS, conflict-free
            v2f bb;
            bb.x = bcol[(size_t)(k0 + khalf)     * d];     // B[k  ][n]
            bb.y = bcol[(size_t)(k0 + khalf + 1) * d];     // B[k+1][n]
            acc = __builtin_amdgcn_wmma_f32_16x16x4_f32(
                    false, a, false, bb, (short)0, acc, false, false);
        }
        // Epilogue: residual from LDS, store Y, accumulate moments.
        float*    ystrip = Y + (size_t)b * DTOT + off + (size_t)mt * 16 * d;
        const int mbase  = (lane >> 4) << 3;               // +8 for upper lanes
#pragma unroll
        for (int j = 0; j < 8; ++j) {
            const int mrow = mbase + j;                    // 0..15 within strip
            float v = acc[j] + As[mrow * pad + ncol];      // residual
            ystrip[(size_t)mrow * d + ncol] = v;
            s1 += v;
            s2 += v * v;
        }
    }

    // wave32 tree reduction; fixed partial slot per (b, strip, wave) so the
    // whole moment pipeline is deterministic (no float atomics).
    for (int o = 16; o > 0; o >>= 1) {
        s1 += __shfl_xor(s1, o, 32);
        s2 += __shfl_xor(s2, o, 32);
    }
    if (lane == 0) {
        const size_t slot = ((size_t)b * NSTRIP + s) * 8 + w;
        partSum[slot] = s1;                        // waves with no tiles write 0
        partSq [slot] = s2;
    }
}

// Deterministic fixed-order fold of the 224 per-wave partials per sample.
__global__ __launch_bounds__(256) void sat_reduce(
    const float* __restrict__ partSum, const float* __restrict__ partSq,
    float* __restrict__ moments)
{
    int b = blockIdx.x * blockDim.x + threadIdx.x;
    if (b >= BATCH) return;
    float s1 = 0.f, s2 = 0.f;
    const float* ps = partSum + (size_t)b * NPART;
    const float* pq = partSq  + (size_t)b * NPART;
    for (int i = 0; i < NPART; ++i) { s1 += ps[i]; s2 += pq[i]; }
    moments[b]         = s1;
    moments[BATCH + b] = s2;
}

// In-place LayerNorm: y = (y - mu) * rsqrt(var + eps) * gamma + beta
__global__ __launch_bounds__(256) void sat_norm(
    float* __restrict__ Y, const float* __restrict__ moments,
    const float* __restrict__ gamma, const float* __restrict__ beta)
{
    const int DQ  = DTOT / 4;                      // 21504 float4 per row
    int tid = blockIdx.x * blockDim.x + threadIdx.x;
    int b   = tid / DQ;
    int q   = tid - b * DQ;
    if (b >= BATCH) return;

    const float mu  = moments[b] * (1.f / (float)DTOT);
    const float var = moments[BATCH + b] * (1.f / (float)DTOT) - mu * mu;
    const float inv = rsqrtf(var + EPSLN);

    float4       y  = *(float4*)(Y + (size_t)b * DTOT + (size_t)q * 4);
    const float4 g  = *(const float4*)(gamma + (size_t)q * 4);
    const float4 be = *(const float4*)(beta  + (size_t)q * 4);
    y.x = (y.x - mu) * inv * g.x + be.x;
    y.y = (y.y - mu) * inv * g.y + be.y;
    y.z = (y.z - mu) * inv * g.z + be.z;
    y.w = (y.w - mu) * inv * g.w + be.w;
    *(float4*)(Y + (size_t)b * DTOT + (size_t)q * 4) = y;
}

extern "C" void kernel_launch(void* const* d_in, const int* in_sizes, int n_in,
                              void* d_out, int out_size, void* d_ws, size_t ws_size,
                              hipStream_t stream) {
    (void)in_sizes; (void)n_in; (void)out_size; (void)ws_size;
    const float* rep0   = (const float*)d_in[0];
    const float* rep1   = (const float*)d_in[1];
    const float* rep2   = (const float*)d_in[2];
    const float* gamma0 = (const float*)d_in[3];
    const float* beta0  = (const float*)d_in[4];
    const float* gamma1 = (const float*)d_in[5];
    const float* beta1  = (const float*)d_in[6];
    const float* gamma2 = (const float*)d_in[7];
    const float* beta2  = (const float*)d_in[8];
    const int*   trans  = (const int*)d_in[9];
    const int*   seq    = (const int*)d_in[10];
    float*       out    = (float*)d_out;

    // Workspace layout (floats): state buffer | partSum | partSq | moments
    float* bufA    = (float*)d_ws;
    float* partSum = bufA    + (size_t)BATCH * DTOT;
    float* partSq  = partSum + (size_t)BATCH * NPART;
    float* moments = partSq  + (size_t)BATCH * NPART;

    const dim3 blk(256);
    const dim3 gMM(BATCH * NSTRIP);                // one block per 16-row strip
    const dim3 gNorm((BATCH * (DTOT / 4)) / 256);
    const dim3 gRed((BATCH + 255) / 256);

    // Layer 0: A/residual from rep[0] directly, write Y -> d_out
    sat_mm    <<<gMM,   blk, 0, stream>>>(rep0, rep1, rep2, trans, seq,
                                          nullptr, out, partSum, partSq, 1);
    sat_reduce<<<gRed,  blk, 0, stream>>>(partSum, partSq, moments);
    sat_norm  <<<gNorm, blk, 0, stream>>>(out, moments, gamma0, beta0);

    // Layer 1: d_out -> bufA
    sat_mm    <<<gMM,   blk, 0, stream>>>(rep0, rep1, rep2, trans, seq,
                                          out, bufA, partSum, partSq, 0);
    sat_reduce<<<gRed,  blk, 0, stream>>>(partSum, partSq, moments);
    sat_norm  <<<gNorm, blk, 0, stream>>>(bufA, moments, gamma1, beta1);

    // Layer 2: bufA -> d_out
    sat_mm    <<<gMM,   blk, 0, stream>>>(rep0, rep1, rep2, trans, seq,
                                          bufA, out, partSum, partSq, 0);
    sat_reduce<<<gRed,  blk, 0, stream>>>(partSum, partSq, moments);
    sat_norm  <<<gNorm, blk, 0, stream>>>(out, moments, gamma2, beta2);
}